// GATLayer_33389075759328
// MI455X (gfx1250) — compile-verified
//
#include <hip/hip_runtime.h>
#include <math.h>

#define N_NODES 50000
#define N_EDGES 800000
#define IN_DIM 128
#define OUT_DIM 128
#define HEADS 8
#define CPH 16            // channels per head
#define NEG_SLOPE 0.2f
#define GAT_EPS 1e-16f

typedef float v2f __attribute__((ext_vector_type(2)));
typedef float v8f __attribute__((ext_vector_type(8)));

// ---------------------------------------------------------------------------
// Kernel 1: h = x @ W  via V_WMMA_F32_16X16X4_F32 (fp32-exact, CDNA5 matrix op)
// Grid: 3125 blocks (50000/16 M-tiles), 256 threads = 8 waves.
// Wave w computes the 16x16 tile (mtile, ntile=w); K=128 -> 32 WMMA ops.
// x-tile (16x128) staged in LDS (padded stride vs 64 banks); W streamed from
// global (128x128 = 64KB, L2-resident across all blocks).
// ---------------------------------------------------------------------------
#define LDX 132  // padded LDS stride (floats) to avoid 64-bank conflicts

__global__ __launch_bounds__(256) void gat_gemm_wmma_kernel(
    const float* __restrict__ x, const float* __restrict__ Wg,
    float* __restrict__ h) {
  __shared__ float sx[16 * LDX];
  const int tid = threadIdx.x;
  const int mbase = blockIdx.x * 16;  // 3125*16 == 50000 exactly

  // Cooperative load of the 16x128 x-tile: 512 float4s across 256 threads.
  for (int i = tid; i < 16 * 32; i += 256) {
    const int r  = i >> 5;   // row 0..15
    const int c4 = i & 31;   // float4 column
    const float4 v = ((const float4*)(x + (size_t)(mbase + r) * IN_DIM))[c4];
    float* dst = &sx[r * LDX + c4 * 4];
    dst[0] = v.x; dst[1] = v.y; dst[2] = v.z; dst[3] = v.w;
  }
  __syncthreads();

  const int wave = tid >> 5;     // ntile 0..7  (head index)
  const int lane = tid & 31;
  const int half = lane >> 4;    // 0: lanes 0-15, 1: lanes 16-31
  const int mr   = lane & 15;    // row within tile / col within tile
  const int nb   = wave * 16;    // output column base

  v8f acc = {};
#pragma unroll
  for (int k0 = 0; k0 < IN_DIM; k0 += 4) {
    // A 16x4 f32 frag: VGPR0 = K {0,2}, VGPR1 = K {1,3} split across halves.
    v2f a, b;
    a.x = sx[mr * LDX + k0 + 2 * half];
    a.y = sx[mr * LDX + k0 + 2 * half + 1];
    // B 4x16 f32 frag (row-striped across lanes), streamed from global/L2.
    b.x = Wg[(size_t)(k0 + 2 * half) * OUT_DIM + nb + mr];
    b.y = Wg[(size_t)(k0 + 2 * half + 1) * OUT_DIM + nb + mr];
    acc = __builtin_amdgcn_wmma_f32_16x16x4_f32(
        /*neg_a=*/false, a, /*neg_b=*/false, b,
        /*c_mod=*/(short)0, acc, /*reuse_a=*/false, /*reuse_b=*/false);
  }

  // D layout: VGPR r -> row M=r (lanes 0-15) / M=r+8 (lanes 16-31), col = lane&15
#pragma unroll
  for (int r = 0; r < 8; ++r) {
    const int row = half * 8 + r;
    h[(size_t)(mbase + row) * OUT_DIM + nb + mr] = acc[r];
  }
}

// ---------------------------------------------------------------------------
// Kernel 2: per-(node,head) attention projections a_src/a_dst
// ---------------------------------------------------------------------------
__global__ void gat_attn_proj_kernel(const float* __restrict__ h,
                                     const float* __restrict__ att_src,
                                     const float* __restrict__ att_dst,
                                     float* __restrict__ a_src,
                                     float* __restrict__ a_dst) {
  const int i = blockIdx.x * blockDim.x + threadIdx.x;  // node*HEADS + head
  if (i >= N_NODES * HEADS) return;
  const int hd = i & (HEADS - 1);
  const float4* hp = (const float4*)(h + (size_t)i * CPH);
  const float4* as = (const float4*)(att_src + hd * CPH);
  const float4* ad = (const float4*)(att_dst + hd * CPH);
  float s = 0.f, d = 0.f;
#pragma unroll
  for (int q = 0; q < 4; ++q) {
    const float4 hv = hp[q], av = as[q], dv = ad[q];
    s += hv.x * av.x + hv.y * av.y + hv.z * av.z + hv.w * av.w;
    d += hv.x * dv.x + hv.y * dv.y + hv.z * dv.z + hv.w * dv.w;
  }
  a_src[i] = s;
  a_dst[i] = d;
}

// ---------------------------------------------------------------------------
// Edge helpers
// ---------------------------------------------------------------------------
__device__ __forceinline__ void edge_nodes(const int* __restrict__ ei, int e,
                                           int& s, int& d) {
  if (e < N_EDGES) { s = ei[e]; d = ei[N_EDGES + e]; }
  else             { s = d = e - N_EDGES; }  // self-loops appended
}
__device__ __forceinline__ float leaky(float v) {
  return v >= 0.f ? v : NEG_SLOPE * v;
}
// Monotonic float <-> uint mapping so segment-max can use hardware atomicMax(u32).
__device__ __forceinline__ unsigned fflip(float f) {
  const unsigned u = __float_as_uint(f);
  return u ^ ((u >> 31) ? 0xFFFFFFFFu : 0x80000000u);
}
__device__ __forceinline__ float funflip(unsigned u) {
  return __uint_as_float(u ^ ((u >> 31) ? 0x80000000u : 0xFFFFFFFFu));
}

#define E2 (N_EDGES + N_NODES)

// Kernel 3: segment max of LeakyReLU(e) over destinations
__global__ void gat_edge_max_kernel(const int* __restrict__ ei,
                                    const float* __restrict__ a_src,
                                    const float* __restrict__ a_dst,
                                    unsigned* __restrict__ emaxU) {
  const int i = blockIdx.x * blockDim.x + threadIdx.x;
  if (i >= E2 * HEADS) return;
  const int e = i >> 3, hd = i & (HEADS - 1);
  int s, d; edge_nodes(ei, e, s, d);
  const float v = leaky(a_src[s * HEADS + hd] + a_dst[d * HEADS + hd]);
  atomicMax(&emaxU[d * HEADS + hd], fflip(v));
}

// Kernel 4: segment sum of exp(e - emax[dst])
__global__ void gat_edge_sum_kernel(const int* __restrict__ ei,
                                    const float* __restrict__ a_src,
                                    const float* __restrict__ a_dst,
                                    const unsigned* __restrict__ emaxU,
                                    float* __restrict__ denom) {
  const int i = blockIdx.x * blockDim.x + threadIdx.x;
  if (i >= E2 * HEADS) return;
  const int e = i >> 3, hd = i & (HEADS - 1);
  int s, d; edge_nodes(ei, e, s, d);
  const int dh = d * HEADS + hd;
  const float v = leaky(a_src[s * HEADS + hd] + a_dst[dh]);
  atomicAdd(&denom[dh], __expf(v - funflip(emaxU[dh])));
}

// Kernel 5: scatter alpha * h[src] into accum[dst] (L2-resident working set)
__global__ void gat_edge_scatter_kernel(const int* __restrict__ ei,
                                        const float* __restrict__ a_src,
                                        const float* __restrict__ a_dst,
                                        const unsigned* __restrict__ emaxU,
                                        const float* __restrict__ denom,
                                        const float* __restrict__ h,
                                        float* __restrict__ accum) {
  const int i = blockIdx.x * blockDim.x + threadIdx.x;  // edge*HEADS + head
  if (i >= E2 * HEADS) return;
  const int e = i >> 3, hd = i & (HEADS - 1);
  int s, d; edge_nodes(ei, e, s, d);
  const int dh = d * HEADS + hd;
  const float v = leaky(a_src[s * HEADS + hd] + a_dst[dh]);
  const float alpha = __expf(v - funflip(emaxU[dh])) / (denom[dh] + GAT_EPS);
  const float4* hs = (const float4*)(h + (size_t)(s * HEADS + hd) * CPH);
  float* op = accum + (size_t)dh * CPH;
#pragma unroll
  for (int q = 0; q < 4; ++q) {
    const float4 hv = hs[q];
    atomicAdd(&op[q * 4 + 0], alpha * hv.x);
    atomicAdd(&op[q * 4 + 1], alpha * hv.y);
    atomicAdd(&op[q * 4 + 2], alpha * hv.z);
    atomicAdd(&op[q * 4 + 3], alpha * hv.w);
  }
}

// Kernel 6: bias + ELU -> d_out
__global__ void gat_finalize_kernel(const float* __restrict__ accum,
                                    const float* __restrict__ bias,
                                    float* __restrict__ out) {
  const int i = blockIdx.x * blockDim.x + threadIdx.x;
  if (i >= N_NODES * OUT_DIM) return;
  const float v = accum[i] + bias[i & (OUT_DIM - 1)];
  out[i] = v > 0.f ? v : expm1f(v);
}

// ---------------------------------------------------------------------------
extern "C" void kernel_launch(void* const* d_in, const int* in_sizes, int n_in,
                              void* d_out, int out_size, void* d_ws, size_t ws_size,
                              hipStream_t stream) {
  (void)in_sizes; (void)n_in; (void)out_size; (void)ws_size;
  const float* x       = (const float*)d_in[0];
  const int*   ei      = (const int*)d_in[1];
  const float* W       = (const float*)d_in[2];
  const float* att_src = (const float*)d_in[3];
  const float* att_dst = (const float*)d_in[4];
  const float* bias    = (const float*)d_in[5];

  // Workspace layout (floats): h | a_src | a_dst | emaxU | denom | accum  (~58 MB)
  float*    h      = (float*)d_ws;
  float*    a_src  = h + (size_t)N_NODES * OUT_DIM;
  float*    a_dst  = a_src + (size_t)N_NODES * HEADS;
  unsigned* emaxU  = (unsigned*)(a_dst + (size_t)N_NODES * HEADS);
  float*    denom  = (float*)emaxU + (size_t)N_NODES * HEADS;
  float*    accum  = denom + (size_t)N_NODES * HEADS;

  // Zero emaxU + denom + accum in one contiguous async memset (graph-capturable).
  hipMemsetAsync(emaxU, 0,
                 (size_t)(2 * N_NODES * HEADS + (size_t)N_NODES * OUT_DIM) *
                     sizeof(float),
                 stream);

  gat_gemm_wmma_kernel<<<N_NODES / 16, 256, 0, stream>>>(x, W, h);

  gat_attn_proj_kernel<<<(N_NODES * HEADS + 255) / 256, 256, 0, stream>>>(
      h, att_src, att_dst, a_src, a_dst);

  const int ethreads = E2 * HEADS;  // 6.8M
  const int eblocks  = (ethreads + 255) / 256;
  gat_edge_max_kernel<<<eblocks, 256, 0, stream>>>(ei, a_src, a_dst, emaxU);
  gat_edge_sum_kernel<<<eblocks, 256, 0, stream>>>(ei, a_src, a_dst, emaxU, denom);
  gat_edge_scatter_kernel<<<eblocks, 256, 0, stream>>>(ei, a_src, a_dst, emaxU,
                                                       denom, h, accum);

  gat_finalize_kernel<<<(N_NODES * OUT_DIM + 255) / 256, 256, 0, stream>>>(
      accum, bias, (float*)d_out);
}